// PCEN_32384053412449
// MI455X (gfx1250) — compile-verified
//
#include <hip/hip_runtime.h>
#include <stdint.h>

// PCEN: per-channel energy normalization.
// x: [B=32, F=128, T=8000] fp32; params s, alpha, delta, r: [F] (log-space).
// One wave32 per (b,f) row; affine-scan parallelization of the IIR smoother;
// tiles streamed through LDS with gfx1250 async global->LDS loads.

#define T_LEN   8000
#define F_CH    128
#define LT      50            // elements per lane per tile
#define TILE    (32 * LT)     // 1600 floats per tile
#define NTILES  (T_LEN / TILE)// 5
#define WPB     4             // waves per block
#define BLOCK   (WPB * 32)
#define EPSV    1e-6f
#define L2EPS   (-19.931568569324174f)   // log2(1e-6)

// Hardware transcendentals (v_exp_f32 = 2^x, v_log_f32 = log2 x)
#define EXP2F(x) __builtin_amdgcn_exp2f(x)
#define LOG2F(x) __builtin_amdgcn_logf(x)

__device__ __forceinline__ void async_load_b64(uint32_t lds_off, const float* g) {
    // GV mode: vdst = LDS byte offset (workgroup-relative), vaddr = 64-bit global addr.
    asm volatile("global_load_async_to_lds_b64 %0, %1, off"
                 :: "v"(lds_off), "v"(g)
                 : "memory");
}
__device__ __forceinline__ void wait_async_le25() {
    asm volatile("s_wait_asynccnt 0x19" ::: "memory");
}
__device__ __forceinline__ void wait_async_0() {
    asm volatile("s_wait_asynccnt 0x0" ::: "memory");
}
__device__ __forceinline__ void wait_ds_0() {
    asm volatile("s_wait_dscnt 0x0" ::: "memory");
}

__global__ __launch_bounds__(BLOCK)
void pcen_scan_kernel(const float* __restrict__ x,
                      const float* __restrict__ s_log,
                      const float* __restrict__ alpha_log,
                      const float* __restrict__ delta_log,
                      const float* __restrict__ r_log,
                      float* __restrict__ out,
                      int rows) {
    __shared__ __align__(16) float smem[WPB * 2 * TILE];

    const int wave = threadIdx.x >> 5;
    const int lane = threadIdx.x & 31;
    const int row  = blockIdx.x * WPB + wave;
    if (row >= rows) return;                 // wave-uniform exit; no barriers used
    const int f    = row & (F_CH - 1);

    const float* xr = x   + (size_t)row * T_LEN;
    float*       yr = out + (size_t)row * T_LEN;

    float* wbuf = smem + wave * (2 * TILE);
    const uint32_t lds_base = (uint32_t)(uintptr_t)(void*)wbuf; // low 32 bits of flat LDS addr = LDS offset

    // Kick off tile-0 prefetch immediately; param transcendentals overlap it.
    #pragma unroll
    for (int j = 0; j < 25; ++j) {
        const int e = j * 64 + lane * 2;
        async_load_b64(lds_base + (uint32_t)(e * 4), xr + e);
    }

    // Per-channel parameters (log-space -> linear): exp(p) = 2^(p*log2(e))
    const float L2E = 1.4426950408889634f;
    const float s_ = EXP2F(s_log[f] * L2E);
    const float a  = 1.0f - s_;
    const float al = EXP2F(alpha_log[f] * L2E);
    const float de = EXP2F(delta_log[f] * L2E);
    const float rr = EXP2F(r_log[f] * L2E);
    const float dr = EXP2F(rr * LOG2F(de));   // delta^r (delta = exp(.) > 0)

    // a^LT by repeated squaring (LT = 50 = 32 + 16 + 2); robust for any sign of a
    const float a2 = a * a, a4 = a2 * a2, a8 = a4 * a4, a16 = a8 * a8, a32 = a16 * a16;
    const float Alane = a32 * a16 * a2;

    // Uniform recurrence m_t = a*m_{t-1} + s*x_t with virtual m_{-1} = x_0
    // reproduces m_0 = x_0 exactly (a*x0 + s*x0 = x0).
    float carry = xr[0];

    for (int t = 0; t < NTILES; ++t) {
        const int cur = t & 1;
        const int nxt = cur ^ 1;

        if (t + 1 < NTILES) {
            const float* src = xr + (t + 1) * TILE;
            #pragma unroll
            for (int j = 0; j < 25; ++j) {
                const int e = j * 64 + lane * 2;
                async_load_b64(lds_base + (uint32_t)((nxt * TILE + e) * 4), src + e);
            }
            wait_async_le25();   // in-order completion => tile t's 25 loads are done
        } else {
            wait_async_0();
        }

        float* buf = wbuf + cur * TILE;
        float* ch  = buf + lane * LT;

        // ---- pass 1: per-lane affine reduction with m_in = 0 ----
        float v = 0.0f;
        #pragma unroll
        for (int i = 0; i < LT; ++i)
            v = fmaf(a, v, s_ * ch[i]);

        // ---- wave-level inclusive scan of affine maps (A, v) ----
        float Aacc = Alane, vacc = v;
        #pragma unroll
        for (int off = 1; off < 32; off <<= 1) {
            const float Ap = __shfl_up(Aacc, off);
            const float vp = __shfl_up(vacc, off);
            if (lane >= off) {
                vacc = fmaf(Aacc, vp, vacc);   // compose: cur ∘ prev
                Aacc *= Ap;
            }
        }
        const float Ae = __shfl_up(Aacc, 1);
        const float ve = __shfl_up(vacc, 1);
        float m = (lane == 0) ? carry : fmaf(Ae, carry, ve);   // exclusive prefix applied to carry
        const float Af = __shfl(Aacc, 31);
        const float vf = __shfl(vacc, 31);
        carry = fmaf(Af, carry, vf);                           // tile-level carry

        // ---- pass 2: regenerate m exactly; pointwise PCEN transform ----
        for (int i = 0; i < LT; ++i) {
            const float xv = ch[i];
            m = fmaf(a, m, s_ * xv);
            const float u    = fmaf(m, 1.0f / EPSV, 1.0f);         // 1 + m/eps
            const float comp = EXP2F(-al * (LOG2F(u) + L2EPS));    // (eps+m)^(-alpha)
            const float base = fmaf(xv, comp, de);                 // > 0
            const float y    = EXP2F(rr * LOG2F(base)) - dr;       // base^r - delta^r
            ch[i] = y;   // stage result in LDS for coalesced store
        }

        // ---- coalesced float2 store of this tile ----
        float* dst = yr + t * TILE;
        #pragma unroll
        for (int j = 0; j < 25; ++j) {
            const int e = j * 64 + lane * 2;
            const float2 val = *(const float2*)(buf + e);
            *(float2*)(dst + e) = val;
        }
        wait_ds_0();   // LDS reads done before async engine refills this buffer
    }
}

extern "C" void kernel_launch(void* const* d_in, const int* in_sizes, int n_in,
                              void* d_out, int out_size, void* d_ws, size_t ws_size,
                              hipStream_t stream) {
    (void)n_in; (void)d_ws; (void)ws_size; (void)out_size;
    const float* x     = (const float*)d_in[0];
    const float* s     = (const float*)d_in[1];
    const float* alpha = (const float*)d_in[2];
    const float* delta = (const float*)d_in[3];
    const float* r     = (const float*)d_in[4];
    float* out = (float*)d_out;

    const int rows = in_sizes[0] / T_LEN;          // B*F = 4096
    const int grid = (rows + WPB - 1) / WPB;       // 1024 blocks of 4 waves
    pcen_scan_kernel<<<grid, BLOCK, 0, stream>>>(x, s, alpha, delta, r, out, rows);
}